// TmapDecoder_14285061226783
// MI455X (gfx1250) — compile-verified
//
#include <hip/hip_runtime.h>
#include <math.h>

// ---- problem geometry (from setup_inputs): disp is [8, 384, 1280, 1] f32 ----
constexpr int Bc = 8, Hc = 384, Wc = 1280;
constexpr int TW = 64, TH = 16;           // output tile per workgroup (exact: 20x24 tiles)
constexpr int PW = TW + 4, PH = TH + 4;   // halo tile for the 5x5 SAME window
constexpr int BDIM = 256;                 // 8 wave32 waves per workgroup

// CDNA5 async global->LDS path (ASYNCcnt-tracked DMA into LDS).
#if defined(__has_builtin)
#if __has_builtin(__builtin_amdgcn_global_load_async_to_lds_b32)
#define USE_ASYNC_LDS 1
#endif
#if __has_builtin(__builtin_amdgcn_wmma_f32_16x16x4_f32)
#define USE_WMMA_K 1
#endif
#endif

#ifdef USE_ASYNC_LDS
// Builtin prototype (per hipcc diagnostic): param0 is global-AS 'int*', param1 LDS-AS 'int*'.
typedef __attribute__((address_space(1))) int* as1_ip;
typedef __attribute__((address_space(3))) int* as3_ip;
#endif

#ifdef USE_WMMA_K
typedef __attribute__((ext_vector_type(2))) float v2f;
typedef __attribute__((ext_vector_type(8))) float v8f;
#endif

// K_scale matrix element (row m, col k): [[rw,0,.5(rw-1)],[0,rh,.5(rh-1)],[0,0,1]]
__device__ __forceinline__ float kscale_elem(int m, int k, float rw, float rh) {
    float v = 0.0f;
    if (m == 0 && k == 0) v = rw;
    else if (m == 0 && k == 2) v = 0.5f * (rw - 1.0f);
    else if (m == 1 && k == 1) v = rh;
    else if (m == 1 && k == 2) v = 0.5f * (rh - 1.0f);
    else if (m == 2 && k == 2) v = 1.0f;
    return v;
}

__global__ __launch_bounds__(BDIM)
void plane_depth_kernel(const float* __restrict__ disp,
                        const float* __restrict__ K0,
                        const float* __restrict__ baseline,
                        const int*   __restrict__ h0p,
                        const int*   __restrict__ w0p,
                        float* __restrict__ out_depth,
                        float* __restrict__ out_mask)
{
    __shared__ float s_x[PH][PW];          // back-projected xyz over the halo tile
    __shared__ float s_y[PH][PW];
    __shared__ float s_z[PH][PW];
    __shared__ float s_h[9][PH][TW];       // horizontal 5-tap sums of 9 moment channels
    __shared__ float s_kpar[4];            // fx, cx, fy, cy of K1 = K_scale @ K0
    float* s_d = &s_h[0][0][0];            // alias: disp staging (PH*PW floats); dead
                                           // before s_h is first written (barrier-separated)

    const int tid = threadIdx.x;
    const int bx = blockIdx.x, by = blockIdx.y, b = blockIdx.z;
    const int gx0 = bx * TW - 2, gy0 = by * TH - 2;

    const float rw = (float)Wc / (float)w0p[0];
    const float rh = (float)Hc / (float)h0p[0];

    // ---- K1 = K_scale @ K0 : a real 3x3 matmul, done on the matrix pipe ----
    // Embedded in V_WMMA_F32_16X16X4_F32 (M=N=16, K=4; 3x3 in the top-left corner).
#ifdef USE_WMMA_K
    if (tid < 32) {                         // one full wave32, EXEC all ones
        const int lane = tid;
        const int m    = lane & 15;         // A: M index / B: N index
        const int half = lane >> 4;         // K pair selector: {0,1} vs {2,3}
        const int k0 = 2 * half, k1 = 2 * half + 1;
        // A-matrix 16x4 f32 layout: VGPR0 = K(k0), VGPR1 = K(k1), lanes 0-15 M=0..15
        v2f a;
        a.x = kscale_elem(m, k0, rw, rh);
        a.y = kscale_elem(m, k1, rw, rh);
        // B-matrix 4x16 f32 layout (mirror): VGPR0 = row K(k0), VGPR1 = row K(k1), N=lane&15
        v2f bm;
        bm.x = (k0 < 3 && m < 3) ? K0[b * 9 + k0 * 3 + m] : 0.0f;
        bm.y = (k1 < 3 && m < 3) ? K0[b * 9 + k1 * 3 + m] : 0.0f;
        v8f c = {};
        c = __builtin_amdgcn_wmma_f32_16x16x4_f32(
                /*neg_a=*/false, a, /*neg_b=*/false, bm,
                /*c_mod=*/(short)0, c, /*reuse_a=*/false, /*reuse_b=*/false);
        // D layout: VGPR r -> lanes 0-15 (M=r, N=lane), lanes 16-31 (M=r+8, N=lane-16)
        if (lane == 0) s_kpar[0] = c[0];                    // K1[0][0] = fx
        if (lane == 2) { s_kpar[1] = c[0]; s_kpar[3] = c[1]; } // K1[0][2]=cx, K1[1][2]=cy
        if (lane == 1) s_kpar[2] = c[1];                    // K1[1][1] = fy
    }
#else
    if (tid == 0) {
        s_kpar[0] = rw * K0[b * 9 + 0];
        s_kpar[1] = rw * K0[b * 9 + 2] + 0.5f * (rw - 1.0f);
        s_kpar[2] = rh * K0[b * 9 + 4];
        s_kpar[3] = rh * K0[b * 9 + 5] + 0.5f * (rh - 1.0f);
    }
#endif

    // ---- stage disparity halo tile into LDS (async DMA on CDNA5) ----
    const float* dbase = disp + (size_t)b * Hc * Wc;
    for (int i = tid; i < PH * PW; i += BDIM) {
        const int lx = i % PW, ly = i / PW;
        const int gx = gx0 + lx, gy = gy0 + ly;
        if (gx >= 0 && gx < Wc && gy >= 0 && gy < Hc) {
#ifdef USE_ASYNC_LDS
            __builtin_amdgcn_global_load_async_to_lds_b32(
                (as1_ip)(dbase + (size_t)gy * Wc + gx), (as3_ip)(s_d + i),
                /*offset=*/0, /*cpol=*/0);
#else
            s_d[i] = dbase[(size_t)gy * Wc + gx];
#endif
        }
    }
#ifdef USE_ASYNC_LDS
#if __has_builtin(__builtin_amdgcn_s_wait_asynccnt)
    __builtin_amdgcn_s_wait_asynccnt(0);
#else
    asm volatile("s_wait_asynccnt 0" ::: "memory");
#endif
#endif
    __syncthreads();

    const float fx = s_kpar[0], cx = s_kpar[1], fy = s_kpar[2], cy = s_kpar[3];
    const float fxb   = fx * baseline[b];
    const float invfx = 1.0f / fx, invfy = 1.0f / fy;

    // ---- back-project each halo pixel once (OOB -> 0, matching zero-padded sums) ----
    for (int i = tid; i < PH * PW; i += BDIM) {
        const int lx = i % PW, ly = i / PW;
        const int gx = gx0 + lx, gy = gy0 + ly;
        float X = 0.f, Y = 0.f, Z = 0.f;
        if (gx >= 0 && gx < Wc && gy >= 0 && gy < Hc) {
            const float z = fxb / ((float)Wc * fmaxf(s_d[i], 1e-6f));
            X = ((float)gx - cx) * z * invfx;
            Y = ((float)gy - cy) * z * invfy;
            Z = z;
        }
        s_x[ly][lx] = X; s_y[ly][lx] = Y; s_z[ly][lx] = Z;
    }
    __syncthreads();

    // ---- horizontal 5-tap sums of {x,y,z, xx,xy,xz, yy,yz,zz} ----
    for (int i = tid; i < PH * TW; i += BDIM) {
        const int ox = i % TW, ly = i / TW;
        float sx = 0.f, sy = 0.f, sz = 0.f;
        float sxx = 0.f, sxy = 0.f, sxz = 0.f, syy = 0.f, syz = 0.f, szz = 0.f;
#pragma unroll
        for (int k = 0; k < 5; ++k) {
            const float X = s_x[ly][ox + k];
            const float Y = s_y[ly][ox + k];
            const float Z = s_z[ly][ox + k];
            sx += X; sy += Y; sz += Z;
            sxx += X * X; sxy += X * Y; sxz += X * Z;
            syy += Y * Y; syz += Y * Z; szz += Z * Z;
        }
        s_h[0][ly][ox] = sx;  s_h[1][ly][ox] = sy;  s_h[2][ly][ox] = sz;
        s_h[3][ly][ox] = sxx; s_h[4][ly][ox] = sxy; s_h[5][ly][ox] = sxz;
        s_h[6][ly][ox] = syy; s_h[7][ly][ox] = syz; s_h[8][ly][ox] = szz;
    }
    __syncthreads();

    // ---- vertical 5-tap sums + covariance + closed-form 3x3 eigen + outputs ----
    for (int i = tid; i < TH * TW; i += BDIM) {
        const int ox = i % TW, oy = i / TW;
        float m0 = 0.f, m1 = 0.f, m2 = 0.f, m3 = 0.f, m4 = 0.f,
              m5 = 0.f, m6 = 0.f, m7 = 0.f, m8 = 0.f;
#pragma unroll
        for (int k = 0; k < 5; ++k) {
            m0 += s_h[0][oy + k][ox]; m1 += s_h[1][oy + k][ox]; m2 += s_h[2][oy + k][ox];
            m3 += s_h[3][oy + k][ox]; m4 += s_h[4][oy + k][ox]; m5 += s_h[5][oy + k][ox];
            m6 += s_h[6][oy + k][ox]; m7 += s_h[7][oy + k][ox]; m8 += s_h[8][oy + k][ox];
        }
        const int gx = bx * TW + ox;
        const int gy = by * TH + oy;
        // SAME-padding valid-element count for this window position
        const int cw = min(gx + 2, Wc - 1) - max(gx - 2, 0) + 1;
        const int ch = min(gy + 2, Hc - 1) - max(gy - 2, 0) + 1;
        const float invc = 1.0f / (float)(cw * ch);

        const float Px = m0 * invc, Py = m1 * invc, Pz = m2 * invc;
        float a00 = m3 * invc - Px * Px;
        float a01 = m4 * invc - Px * Py;
        float a02 = m5 * invc - Px * Pz;
        float a11 = m6 * invc - Py * Py;
        float a12 = m7 * invc - Py * Pz;
        float a22 = m8 * invc - Pz * Pz;

        float A_max = fmaxf(fabsf(a00), fabsf(a01));
        A_max = fmaxf(A_max, fabsf(a02));
        A_max = fmaxf(A_max, fabsf(a11));
        A_max = fmaxf(A_max, fabsf(a12));
        A_max = fmaxf(A_max, fabsf(a22));
        const float eval_zero = (A_max < 1e-6f) ? 1.0f : 0.0f;
        const float sc = 1.0f / fmaxf(A_max, 1e-6f);
        a00 *= sc; a01 *= sc; a02 *= sc; a11 *= sc; a12 *= sc; a22 *= sc;

        const float nrm = a01 * a01 + a02 * a02 + a12 * a12;
        const float q   = (a00 + a11 + a22) * (1.0f / 3.0f);
        const float b00 = a00 - q, b11 = a11 - q, b22 = a22 - q;
        const float p = sqrtf((b00 * b00 + b11 * b11 + b22 * b22 + 2.0f * nrm) * (1.0f / 6.0f));
        const float eig_triple = ((p < 1e-6f) ? 1.0f : 0.0f) * (1.0f - eval_zero);
        const float c00 = b11 * b22 - a12 * a12;
        const float c01 = a01 * b22 - a12 * a02;
        const float c02 = a01 * a12 - b11 * a02;
        const float pm = fmaxf(p, 1e-6f);
        const float det = (b00 * c00 - a01 * c01 + a02 * c02) / (pm * pm * pm);
        const float half_det = fminf(fmaxf(0.5f * det, -1.0f), 1.0f);
        const float angle = acosf(half_det) * (1.0f / 3.0f);
        const float beta2 = 2.0f * cosf(angle);
        const float beta0 = 2.0f * cosf(angle + 2.0943951023931953f); // + 2*pi/3
        const float beta1 = -(beta0 + beta2);
        const float wsel = (1.0f - eval_zero) * (1.0f - eig_triple);
        const float e0 = eig_triple * q + wsel * (q + p * beta0);
        const float e1 = eig_triple * q + wsel * (q + p * beta1);
        const float e2 = eig_triple * q + wsel * (q + p * beta2);
        const float denom = sqrtf(fmaxf((e0 - e1) * (e0 - e2), 0.0f));
        const float evec_zero = (denom < 1e-6f) ? 1.0f : 0.0f;
        const float ny_num = sqrtf(fmaxf(e0 * e0 - (a00 + a22) * e0 + (a00 * a22 - a02 * a02), 0.0f));
        const float ny = fminf(fmaxf(ny_num / fmaxf(denom, 1e-6f), 0.0f), 1.0f);
        const float plane_xz = (1.0f - evec_zero) * ny;

        const float Ycen = s_y[oy + 2][ox + 2];
        const float Zcen = s_z[oy + 2][ox + 2];
        const float mask = ((Ycen > 0.3f) ? 1.0f : 0.0f) * ((plane_xz > 0.85f) ? 1.0f : 0.0f);

        const size_t gi = ((size_t)b * Hc + gy) * Wc + gx;
        out_depth[gi] = Zcen;
        out_mask[gi]  = mask;
    }
}

extern "C" void kernel_launch(void* const* d_in, const int* in_sizes, int n_in,
                              void* d_out, int out_size, void* d_ws, size_t ws_size,
                              hipStream_t stream) {
    (void)in_sizes; (void)n_in; (void)out_size; (void)d_ws; (void)ws_size;
    const float* disp     = (const float*)d_in[0];
    const float* K0       = (const float*)d_in[1];
    const float* baseline = (const float*)d_in[2];
    const int*   h0       = (const int*)d_in[3];
    const int*   w0       = (const int*)d_in[4];
    float* out       = (float*)d_out;
    float* out_depth = out;                             // [B,H,W,1] flat
    float* out_mask  = out + (size_t)Bc * Hc * Wc;      // [B,H,W]   flat

    dim3 grid(Wc / TW, Hc / TH, Bc);   // 20 x 24 x 8
    hipLaunchKernelGGL(plane_depth_kernel, grid, dim3(BDIM), 0, stream,
                       disp, K0, baseline, h0, w0, out_depth, out_mask);
}